// GcnModel_57578331570298
// MI455X (gfx1250) — compile-verified
//
#include <hip/hip_runtime.h>

#define N_FEAT 128

typedef __attribute__((ext_vector_type(2))) float v2f;
typedef __attribute__((ext_vector_type(8))) float v8f;

// ---------------------------------------------------------------- utilities
__global__ void zero_f32(float* __restrict__ p, int n) {
  int i = blockIdx.x * blockDim.x + threadIdx.x;
  if (i < n) p[i] = 0.0f;
}

__global__ void degree_kernel(const int* __restrict__ src, const int* __restrict__ dst,
                              float* __restrict__ outdeg, float* __restrict__ indeg, int E) {
  int e = blockIdx.x * blockDim.x + threadIdx.x;
  if (e < E) {
    atomicAdd(&outdeg[src[e]], 1.0f);
    atomicAdd(&indeg[dst[e]], 1.0f);
  }
}

__global__ void norm_kernel(float* __restrict__ ns, float* __restrict__ nd, int n) {
  int i = blockIdx.x * blockDim.x + threadIdx.x;
  if (i < n) {
    ns[i] = rsqrtf(fmaxf(ns[i], 1.0f));
    nd[i] = rsqrtf(fmaxf(nd[i], 1.0f));
  }
}

// One wave32 per edge: msum[dst] += x[src] * ns[src], 128 floats = float4/lane.
__global__ void scatter_kernel(const float* __restrict__ x, const float* __restrict__ ns,
                               const int* __restrict__ src, const int* __restrict__ dst,
                               float* __restrict__ msum, int E) {
  int gid = blockIdx.x * blockDim.x + threadIdx.x;
  int e = gid >> 5;
  int lane = gid & 31;
  if (e >= E) return;
  int s = src[e];
  int d = dst[e];
  float sc = ns[s];
  float4 v = *(const float4*)(x + (size_t)s * N_FEAT + lane * 4);
  float* o = msum + (size_t)d * N_FEAT + lane * 4;
  atomicAdd(o + 0, v.x * sc);
  atomicAdd(o + 1, v.y * sc);
  atomicAdd(o + 2, v.z * sc);
  atomicAdd(o + 3, v.w * sc);
}

// ---------------------------------------------------------------- fused GEMM
// out[r, :] = LN?ReLU( (msum[r,:] * nd[r]) @ W + bias )
// 256 threads = 8 waves; each wave owns a 16-row band and all DOUT columns,
// computed as DOUT/16 accumulator tiles via V_WMMA_F32_16X16X4_F32 (K-loop of 32).
// A layout: lanes 0-15 hold rows 0..15, k={k0,k0+1}; lanes 16-31 rows 0..15, k={k0+2,k0+3}.
// B layout (mirror): VGPR j at lane l holds W[k0 + 2*(l/16) + j, n = l%16].
// W is staged in LDS packed as k-pair float2 so each lane's B fragment is one
// aligned ds_load_b64 straight into the WMMA source register pair (no re-pairing
// movs, conflict-free: 32 lanes x 8B = one full 64-bank sweep).
// C/D layout: VGPR v at lane l holds out[row v + 8*(l/16), col l%16].
template <int DOUT, bool FUSE_LN>
__global__ __launch_bounds__(256) void gcn_gemm_kernel(
    const float* __restrict__ M,    // [nrows, 128] scatter sums
    const float* __restrict__ ndv,  // [nrows] dst-side degree norms
    const float* __restrict__ W,    // [128, DOUT] row-major
    const float* __restrict__ bias, // [DOUT]
    const float* __restrict__ gamma,
    const float* __restrict__ beta,
    float* __restrict__ out,        // [nrows, DOUT]
    int nrows) {
  constexpr int NT = DOUT / 16;             // column tiles per wave
  constexpr int NPAIR = (N_FEAT / 2) * DOUT;
  __shared__ v2f sWp[NPAIR];                // sWp[(k/2)*DOUT + n] = {W[k,n], W[k+1,n]}

  for (int i = threadIdx.x; i < NPAIR; i += 256) {
    int kk = i / DOUT;       // k-pair index
    int n = i - kk * DOUT;   // column
    v2f w;
    w[0] = W[(2 * kk + 0) * DOUT + n];
    w[1] = W[(2 * kk + 1) * DOUT + n];
    sWp[i] = w;
  }
  __syncthreads();

  const int wave = threadIdx.x >> 5;
  const int lane = threadIdx.x & 31;
  const int g = lane >> 4;   // half-wave group
  const int lr = lane & 15;

  const int row0 = blockIdx.x * 128 + wave * 16;

  // A-fragment source row for this lane (branch-free tail masking: EXEC must
  // stay all-ones through the WMMA loop).
  const int a_row = row0 + lr;
  const int ar = a_row < nrows ? a_row : (nrows - 1);
  const float mask = a_row < nrows ? 1.0f : 0.0f;
  const float ndr = ndv[ar] * mask;
  const float* __restrict__ arow = M + (size_t)ar * N_FEAT;

  v8f acc[NT] = {};

#pragma unroll 8
  for (int ks = 0; ks < N_FEAT / 4; ++ks) {
    const int k0 = ks * 4 + g * 2;
    v2f a;
    a[0] = arow[k0 + 0] * ndr;
    a[1] = arow[k0 + 1] * ndr;
    const v2f* __restrict__ brow = sWp + (2 * ks + g) * DOUT + lr;
#pragma unroll
    for (int t = 0; t < NT; ++t) {
      v2f b = brow[t * 16];
      acc[t] = __builtin_amdgcn_wmma_f32_16x16x4_f32(
          /*neg_a=*/false, a, /*neg_b=*/false, b,
          /*c_mod=*/(short)0, acc[t], /*reuse_a=*/false, /*reuse_b=*/false);
    }
  }

  // ------------------------------------------------------------- epilogue
  float bc[NT];
  float gc[NT];
  float bec[NT];
#pragma unroll
  for (int t = 0; t < NT; ++t) {
    bc[t] = bias[t * 16 + lr];
    if constexpr (FUSE_LN) {
      gc[t] = gamma[t * 16 + lr];
      bec[t] = beta[t * 16 + lr];
    } else {
      gc[t] = 0.0f;
      bec[t] = 0.0f;
    }
  }

#pragma unroll
  for (int t = 0; t < NT; ++t)
#pragma unroll
    for (int v = 0; v < 8; ++v) acc[t][v] += bc[t];

  if constexpr (FUSE_LN) {
    // Row r = v + 8*g lives in acc[*][v] across the 16 lanes of group g.
#pragma unroll
    for (int v = 0; v < 8; ++v) {
      float s = 0.0f, ss = 0.0f;
#pragma unroll
      for (int t = 0; t < NT; ++t) {
        float x = acc[t][v];
        s += x;
        ss += x * x;
      }
#pragma unroll
      for (int off = 1; off < 16; off <<= 1) {
        s += __shfl_xor(s, off, 32);
        ss += __shfl_xor(ss, off, 32);
      }
      const float mu = s * (1.0f / DOUT);
      const float var = ss * (1.0f / DOUT) - mu * mu;
      const float rs = rsqrtf(var + 1e-5f);
#pragma unroll
      for (int t = 0; t < NT; ++t) {
        float y = (acc[t][v] - mu) * rs * gc[t] + bec[t];
        acc[t][v] = fmaxf(y, 0.0f);
      }
    }
  }

#pragma unroll
  for (int v = 0; v < 8; ++v) {
    const int row = row0 + v + 8 * g;
    if (row < nrows) {
      float* __restrict__ orow = out + (size_t)row * DOUT;
#pragma unroll
      for (int t = 0; t < NT; ++t) orow[t * 16 + lr] = acc[t][v];
    }
  }
}

// ---------------------------------------------------------------- launcher
extern "C" void kernel_launch(void* const* d_in, const int* in_sizes, int n_in,
                              void* d_out, int out_size, void* d_ws, size_t ws_size,
                              hipStream_t stream) {
  (void)n_in; (void)out_size; (void)ws_size;

  const float* h   = (const float*)d_in[0];
  const int*   src = (const int*)d_in[1];
  const int*   dst = (const int*)d_in[2];
  const float* W1  = (const float*)d_in[3];
  const float* b1  = (const float*)d_in[4];
  const float* g1  = (const float*)d_in[5];
  const float* be1 = (const float*)d_in[6];
  const float* W2  = (const float*)d_in[7];
  const float* b2  = (const float*)d_in[8];
  const float* g2  = (const float*)d_in[9];
  const float* be2 = (const float*)d_in[10];
  const float* W3  = (const float*)d_in[11];
  const float* b3  = (const float*)d_in[12];

  const int N = in_sizes[0] / N_FEAT;
  const int E = in_sizes[1];

  // Workspace layout: ns[N] | nd[N] | msum[N*128] | hbuf[N*128]
  float* ns   = (float*)d_ws;
  float* nd   = ns + N;
  float* msum = nd + N;
  float* hbuf = msum + (size_t)N * N_FEAT;
  float* out  = (float*)d_out;

  const int feat_elems = N * N_FEAT;
  const dim3 blk(256);
  const dim3 grid_deg((E + 255) / 256);
  const dim3 grid_node((N + 255) / 256);
  const dim3 grid_zero2N((2 * N + 255) / 256);
  const dim3 grid_zeroF((feat_elems + 255) / 256);
  const dim3 grid_scatter((E + 7) / 8);       // 8 edges (waves) per 256-thread block
  const dim3 grid_gemm((N + 127) / 128);      // 8 waves x 16 rows per block

  // degree norms
  zero_f32<<<grid_zero2N, blk, 0, stream>>>(ns, 2 * N);
  degree_kernel<<<grid_deg, blk, 0, stream>>>(src, dst, ns, nd, E);
  norm_kernel<<<grid_node, blk, 0, stream>>>(ns, nd, N);

  // layer 1: h -> hbuf (GraphConv + LN + ReLU)
  zero_f32<<<grid_zeroF, blk, 0, stream>>>(msum, feat_elems);
  scatter_kernel<<<grid_scatter, blk, 0, stream>>>(h, ns, src, dst, msum, E);
  gcn_gemm_kernel<128, true><<<grid_gemm, blk, 0, stream>>>(
      msum, nd, W1, b1, g1, be1, hbuf, N);

  // layer 2: hbuf -> hbuf (scatter finishes reading before gemm rewrites)
  zero_f32<<<grid_zeroF, blk, 0, stream>>>(msum, feat_elems);
  scatter_kernel<<<grid_scatter, blk, 0, stream>>>(hbuf, ns, src, dst, msum, E);
  gcn_gemm_kernel<128, true><<<grid_gemm, blk, 0, stream>>>(
      msum, nd, W2, b2, g2, be2, hbuf, N);

  // layer 3: hbuf -> out (GraphConv only, d_out = 64)
  zero_f32<<<grid_zeroF, blk, 0, stream>>>(msum, feat_elems);
  scatter_kernel<<<grid_scatter, blk, 0, stream>>>(hbuf, ns, src, dst, msum, E);
  gcn_gemm_kernel<64, false><<<grid_gemm, blk, 0, stream>>>(
      msum, nd, W3, b3, nullptr, nullptr, out, N);
}